// InvGridSamplerDenominator_65712999629447
// MI455X (gfx1250) — compile-verified
//
#include <hip/hip_runtime.h>

// Problem constants (from reference setup_inputs):
//   x: (8,16,512,512) f32  -- shape only, values unused
//   inv_grid: (8,512,512,2) f32
//   out: (8,16,512,512) f32
#define BS 8
#define NC 16
#define HH 512
#define WW 512
#define H3 515
#define W3 515

typedef float v2f __attribute__((ext_vector_type(2)));
typedef float v8f __attribute__((ext_vector_type(8)));

// ---------------------------------------------------------------------------
// Kernel 1: bilinear splat scatter-add into per-batch (H3 x W3) accumulator.
// 1 thread per grid point; 4 hardware float atomics into L2-resident buffer.
// ---------------------------------------------------------------------------
__global__ void __launch_bounds__(256)
splat_kernel(const float* __restrict__ grid, float* __restrict__ acc) {
    const int t = blockIdx.x * blockDim.x + threadIdx.x;   // 0 .. BS*HH*WW-1
    const float2 v = ((const float2*)grid)[t];
    const int b = t >> 18;                                  // HH*WW = 262144

    // g = (inv_grid + 1) * 0.5 ; coords = g*dim + 1, clamped.
    // fp32(513 - 2e-10) == 513.0f, so the upper clamp is exactly 513.0f.
    float g0 = (v.x + 1.0f) * 0.5f;
    float g1 = (v.y + 1.0f) * 0.5f;
    float gi = fminf(fmaxf(g0 * 512.0f + 1.0f, 0.0f), 513.0f);
    float gj = fminf(fmaxf(g1 * 512.0f + 1.0f, 0.0f), 513.0f);
    int ii = (int)gi;                // floor (non-negative)
    int jj = (int)gj;
    float fi = gi - (float)ii;       // exact (Sterbenz); relu weights = (1-f, f)
    float fj = gj - (float)jj;
    float wi0 = 1.0f - fi, wi1 = fi;
    float wj0 = 1.0f - fj, wj1 = fj;

    float* base = acc + ((size_t)b * H3 + ii) * W3 + jj;
    unsafeAtomicAdd(base,          wi0 * wj0);   // -> global_atomic_add_f32
    unsafeAtomicAdd(base + 1,      wi0 * wj1);
    unsafeAtomicAdd(base + W3,     wi1 * wj0);
    unsafeAtomicAdd(base + W3 + 1, wi1 * wj1);
}

// ---------------------------------------------------------------------------
// Kernel 2: crop interior + broadcast across 16 channels via WMMA.
// out[b, 0:16, i, j:j+16] = ones(16) (outer) B[b, i+1, j+1:j+17]
// One wave per 16-pixel tile: D(16x16) = A(16x4, col0 = ones) x B(4x16, row0 = den).
// A layout (32-bit A 16x4): VGPR0 lanes 0-15 hold K=0 -> a.x = (lane<16).
// Rows K=1..3 of B are don't-care since A zeroes them.
// ---------------------------------------------------------------------------
__global__ void __launch_bounds__(256)
bcast_wmma_kernel(const float* __restrict__ acc, float* __restrict__ out) {
    const int lane = threadIdx.x & 31;
    const int wave = (blockIdx.x * blockDim.x + threadIdx.x) >> 5;

    const int tpr = WW / 16;                  // 32 tiles per row
    const int b   = wave / (HH * tpr);
    const int rem = wave % (HH * tpr);
    const int i   = rem / tpr;
    const int j   = (rem % tpr) * 16;
    const int col = lane & 15;

    // Interior crop: out[b,c,i,j+col] = acc[b, i+1, j+1+col]
    const float bval = acc[((size_t)b * H3 + (i + 1)) * W3 + (j + 1 + col)];

    v2f a, bm;
    a.x  = (lane < 16) ? 1.0f : 0.0f;         // A: K=0 column = ones, rest 0
    a.y  = 0.0f;
    bm.x = bval;                              // B: K=0 row = denominators
    bm.y = 0.0f;
    v8f c = {};
    c = __builtin_amdgcn_wmma_f32_16x16x4_f32(
            /*neg_a=*/false, a, /*neg_b=*/false, bm,
            /*c_mod=*/(short)0, c, /*reuse_a=*/false, /*reuse_b=*/false);

    // D layout: VGPR v -> channel v (lanes 0-15) / channel v+8 (lanes 16-31),
    // pixel column = lane & 15. 8 stores with immediate offsets v*HW (<= 7 MiB).
    const int choff = (lane < 16) ? 0 : 8;
    const size_t HW = (size_t)HH * WW;
    float* p = out + (((size_t)b * NC + choff) * HH + i) * WW + j + col;
#pragma unroll
    for (int v = 0; v < 8; ++v)
        p[v * HW] = c[v];
}

// ---------------------------------------------------------------------------
extern "C" void kernel_launch(void* const* d_in, const int* in_sizes, int n_in,
                              void* d_out, int out_size, void* d_ws, size_t ws_size,
                              hipStream_t stream) {
    // d_in[0] = x (unused, shape only), d_in[1] = inv_grid
    const float* inv_grid = (const float*)d_in[1];
    float* acc = (float*)d_ws;                 // BS*H3*W3 floats ~ 8.1 MiB
    float* out = (float*)d_out;

    const size_t acc_bytes = (size_t)BS * H3 * W3 * sizeof(float);
    hipMemsetAsync(acc, 0, acc_bytes, stream); // capturable memset node

    const int npts = BS * HH * WW;             // 2,097,152 (multiple of 256)
    splat_kernel<<<npts / 256, 256, 0, stream>>>(inv_grid, acc);

    const int nwaves = BS * HH * (WW / 16);    // 131,072 full waves
    bcast_wmma_kernel<<<(nwaves * 32) / 256, 256, 0, stream>>>(acc, out);
}